// GFunc_79800492359773
// MI455X (gfx1250) — compile-verified
//
#include <hip/hip_runtime.h>
#include <cmath>

typedef __attribute__((ext_vector_type(2))) float v2f;
typedef __attribute__((ext_vector_type(8))) float v8f;

#define KDEG 3
#define N_COEF 13      // NUM_GRIDS + K
#define KDIM 16        // augmented K: 13 basis + x + silu(x) + zero pad
#define M_TILE 32      // batch rows per workgroup (fixed strip)
#define N_TILE 64      // output features per inner-loop iteration
#define LDF 18         // padded LDS row stride in floats (8B-aligned, conflict-free)

__global__ __launch_bounds__(256)
void kan_softplus_wmma(const float* __restrict__ x,
                       const float* __restrict__ coef,
                       const float* __restrict__ w_lin,
                       float* __restrict__ out,
                       int nOut)
{
    __shared__ float f_lds[M_TILE * LDF];   // per-row feature vectors (A), loop-invariant
    __shared__ float c_lds[N_TILE * LDF];   // augmented coefficient tile (B)

    const int tid = threadIdx.x;
    const int m0  = blockIdx.x * M_TILE;

    // ---- stage A (once per WG): cubic B-spline basis + x + silu(x) per batch row ----
    if (tid < M_TILE) {
        const float xv = x[m0 + tid];
        float t[17];
        #pragma unroll
        for (int i = 0; i < 17; ++i) t[i] = 0.1f * (float)i - 0.3f;
        float Bv[16];
        #pragma unroll
        for (int j = 0; j < 16; ++j)
            Bv[j] = (xv >= t[j] && xv < t[j + 1]) ? 1.0f : 0.0f;
        #pragma unroll
        for (int p = 1; p <= KDEG; ++p) {
            const float invd = 1.0f / (0.1f * (float)p);
            #pragma unroll
            for (int j = 0; j < 16 - p; ++j)
                Bv[j] = (xv - t[j]) * invd * Bv[j] + (t[j + p + 1] - xv) * invd * Bv[j + 1];
        }
        float* fr = &f_lds[tid * LDF];
        #pragma unroll
        for (int k = 0; k < N_COEF; ++k) fr[k] = Bv[k];
        fr[13] = xv;                                              // linear term feature
        fr[14] = xv / (1.0f + __builtin_amdgcn_exp2f(-xv * 1.44269504f));  // silu feature
        fr[15] = 0.0f;
    }

    // wave/lane decomposition: 8 wave32s arranged 2 (M) x 4 (N) -> 32x64 per iteration
    const int wid  = tid >> 5;
    const int lane = tid & 31;
    const int half = lane >> 4;           // 0: K pair {0,1}, 1: K pair {2,3}
    const int i16  = lane & 15;
    const int m_base = (wid >> 2) << 4;
    const int n_sub  = (wid & 3) << 4;

    __syncthreads();

    // Hoist loop-invariant A fragments into registers (f32 WMMA A: 2 VGPRs per step)
    v2f afrag[4];
    #pragma unroll
    for (int j = 0; j < 4; ++j)
        afrag[j] = *(const v2f*)&f_lds[(m_base + i16) * LDF + 4 * j + 2 * half];

    // Output base pointer for this lane (row m_base+half*8, col n_sub+i16)
    const size_t row0 = (size_t)(m0 + m_base + half * 8);
    float* const obase = out + row0 * (size_t)nOut + (n_sub + i16);

    // ---- loop over all N tiles, reusing features ----
    for (int nt = 0; nt < nOut; nt += N_TILE) {
        __syncthreads();   // previous iteration's c_lds reads complete

        // stage augmented coefficient tile into LDS
        for (int t = tid; t < N_TILE * KDIM; t += 256) {
            const int n = t >> 4;
            const int k = t & 15;
            float v;
            if (k < N_COEF)   v = coef[(size_t)(nt + n) * N_COEF + k];
            else if (k == 13) v = w_lin[nt + n];
            else if (k == 14) v = 1.0f;
            else              v = 0.0f;
            c_lds[n * LDF + k] = v;
        }
        __syncthreads();

        // 16x16 tile per wave via 4 chained f32 WMMAs (K = 16)
        v8f acc = {};
        #pragma unroll
        for (int j = 0; j < 4; ++j) {
            v2f b = *(const v2f*)&c_lds[(n_sub + i16) * LDF + 4 * j + 2 * half];
            acc = __builtin_amdgcn_wmma_f32_16x16x4_f32(false, afrag[j], false, b,
                                                        (short)0, acc, false, false);
        }

        // softplus (stable, raw v_exp_f32/v_log_f32: arg of exp2 is <= 0, no range
        // reduction needed) + non-temporal streaming store
        float* op = obase + nt;
        #pragma unroll
        for (int r = 0; r < 8; ++r) {
            const float z  = acc[r];
            const float e  = __builtin_amdgcn_exp2f(-fabsf(z) * 1.44269504f);
            const float sp = fmaxf(z, 0.0f) + 0.69314718f * __builtin_amdgcn_logf(1.0f + e);
            __builtin_nontemporal_store(sp, op + (size_t)r * (size_t)nOut);
        }
    }
}

extern "C" void kernel_launch(void* const* d_in, const int* in_sizes, int n_in,
                              void* d_out, int out_size, void* d_ws, size_t ws_size,
                              hipStream_t stream) {
    const float* x    = (const float*)d_in[0];   // (BATCH, 1) fp32
    const float* coef = (const float*)d_in[1];   // (OUT, 1, 13) fp32
    const float* wlin = (const float*)d_in[2];   // (OUT, 1) fp32
    float* out = (float*)d_out;                  // (BATCH, OUT) fp32

    const int batch = in_sizes[0];   // 65536
    const int nOut  = in_sizes[2];   // 2048

    dim3 grid(batch / M_TILE);       // 2048 WGs, each owns a 32-row M strip
    kan_softplus_wmma<<<grid, dim3(256), 0, stream>>>(x, coef, wlin, out, nOut);
}